// Model_47742856462354
// MI455X (gfx1250) — compile-verified
//
#include <hip/hip_runtime.h>
#include <hip/hip_bf16.h>

#define NODES   50000
#define EDGES   800000
#define NEDGE   (EDGES + NODES)   // + self loops
#define HEADS   8
#define CH      16
#define FEAT    128
#define NGRAPH  64
#define LN_EPS  1e-5f
#define ASTRIDE 129               // LDS pad: 129 mod 64 == 1 -> conflict-free cols
#define NPB     64                // nodes per block in node_update

typedef float v2f __attribute__((ext_vector_type(2)));
typedef float v8f __attribute__((ext_vector_type(8)));

__device__ __forceinline__ unsigned f2key(float f) {
  unsigned u = __float_as_uint(f);
  return (u & 0x80000000u) ? ~u : (u | 0x80000000u);
}
__device__ __forceinline__ float key2f(unsigned k) {
  unsigned u = (k & 0x80000000u) ? (k ^ 0x80000000u) : ~k;
  return __uint_as_float(u);
}

// ---------------------------------------------------------------------------
// Kernel 1: XP = H @ W (fp32 WMMA 16x16x4), fused attention dots AS/AD.
// One block = 16 rows; wave w (0..7) computes cols [16w,16w+16) == head w.
// ---------------------------------------------------------------------------
__global__ __launch_bounds__(256) void gemm_attn(
    const float* __restrict__ Xin, const float* __restrict__ Wm,
    const float* __restrict__ a_s, const float* __restrict__ a_d,
    float* __restrict__ XP, float* __restrict__ AS, float* __restrict__ AD)
{
  __shared__ float Apan[16 * ASTRIDE];
  const int lane = threadIdx.x & 31;
  const int head = threadIdx.x >> 5;
  const int row0 = blockIdx.x * 16;
  const int l15  = lane & 15;
  const int hi   = lane >> 4;

  for (int i = threadIdx.x; i < 16 * FEAT; i += 256) {
    int r = i >> 7, c = i & (FEAT - 1);
    int gr = row0 + r;
    Apan[r * ASTRIDE + c] = (gr < NODES) ? Xin[gr * FEAT + c] : 0.0f;
  }
  __syncthreads();

  const int n0 = head * 16;
  v8f acc = {};
  // A 16x4 frag: lane l15 = row M, VGPR0/1 = K { 2*hi, 2*hi+1 }
  // B 4x16 frag: lanes 0-15 = K row 2*hi, lanes 16-31 = K row 2*hi (+1 in .y)
  #pragma unroll
  for (int k0 = 0; k0 < FEAT; k0 += 4) {
    const int ka = k0 + hi * 2;
    v2f a, b;
    a.x = Apan[l15 * ASTRIDE + ka];
    a.y = Apan[l15 * ASTRIDE + ka + 1];
    b.x = Wm[ka * FEAT + n0 + l15];
    b.y = Wm[(ka + 1) * FEAT + n0 + l15];
    acc = __builtin_amdgcn_wmma_f32_16x16x4_f32(false, a, false, b,
                                                (short)0, acc, false, false);
  }

  // C/D frag: VGPR r -> (M = r + 8*hi, N = l15)
  const float asv = a_s[head * CH + l15];
  const float adv = a_d[head * CH + l15];
  #pragma unroll
  for (int r = 0; r < 8; ++r) {
    const int gr = row0 + r + hi * 8;
    const float v = acc[r];
    if (gr < NODES) XP[gr * FEAT + n0 + l15] = v;
    float ps = v * asv;
    float pd = v * adv;
    #pragma unroll
    for (int off = 8; off >= 1; off >>= 1) {   // reduce within 16-lane half
      ps += __shfl_xor(ps, off, 32);
      pd += __shfl_xor(pd, off, 32);
    }
    if (l15 == 0 && gr < NODES) {
      AS[gr * HEADS + head] = ps;
      AD[gr * HEADS + head] = pd;
    }
  }
}

// ---------------------------------------------------------------------------
__global__ void fill_u32(unsigned* __restrict__ p, unsigned v, int n) {
  int i = blockIdx.x * blockDim.x + threadIdx.x;
  if (i < n) p[i] = v;
}

// Kernel 2: per-(dst,head) segment max of leaky-relu logits (uint-key atomics)
__global__ void edge_max(const int* __restrict__ ei, const float* __restrict__ AS,
                         const float* __restrict__ AD, unsigned* __restrict__ Mkey)
{
  int e = blockIdx.x * blockDim.x + threadIdx.x;
  if (e >= NEDGE) return;
  int s, d;
  if (e < EDGES) { s = ei[e]; d = ei[EDGES + e]; } else { s = e - EDGES; d = s; }
  #pragma unroll
  for (int h = 0; h < HEADS; ++h) {
    float l = AS[s * HEADS + h] + AD[d * HEADS + h];
    l = (l > 0.0f) ? l : 0.2f * l;
    atomicMax(&Mkey[d * HEADS + h], f2key(l));
  }
}

// Kernel 3: one wave per edge: ex = exp(l - m); denom += ex; agg += xp[src]*ex
__global__ __launch_bounds__(256) void edge_scatter(
    const int* __restrict__ ei, const float* __restrict__ AS,
    const float* __restrict__ AD, const unsigned* __restrict__ Mkey,
    const float* __restrict__ XP, float* __restrict__ denom, float* __restrict__ agg)
{
  const int wid = blockIdx.x * 8 + (threadIdx.x >> 5);
  if (wid >= NEDGE) return;
  const int lane = threadIdx.x & 31;
  int s, d;
  if (wid < EDGES) { s = ei[wid]; d = ei[EDGES + wid]; } else { s = wid - EDGES; d = s; }

  float ex[4];
  #pragma unroll
  for (int j = 0; j < 4; ++j) {
    const int f = lane + 32 * j;
    const int h = f >> 4;
    float l = AS[s * HEADS + h] + AD[d * HEADS + h];
    l = (l > 0.0f) ? l : 0.2f * l;
    ex[j] = __expf(l - key2f(Mkey[d * HEADS + h]));
  }
  if ((lane & 15) == 0) {                      // lane0: heads 0,2,4,6; lane16: 1,3,5,7
    #pragma unroll
    for (int j = 0; j < 4; ++j)
      atomicAdd(&denom[d * HEADS + ((lane >> 4) + 2 * j)], ex[j]);
  }
  #pragma unroll
  for (int j = 0; j < 4; ++j) {
    const int f = lane + 32 * j;
    atomicAdd(&agg[d * FEAT + f], XP[s * FEAT + f] * ex[j]);
  }
}

// Kernel 4: y = prev + agg/denom + bias; partial per-feature sum / sumsq
__global__ __launch_bounds__(128) void node_update(
    const float* __restrict__ agg, const float* __restrict__ denom,
    const float* prev, const float* __restrict__ bias,
    float* Y, float* __restrict__ fsum, float* __restrict__ fsq)
{
  const int f  = threadIdx.x;
  const int n0 = blockIdx.x * NPB;
  float s = 0.0f, sq = 0.0f;
  for (int i = 0; i < NPB; ++i) {
    const int n = n0 + i;
    if (n >= NODES) break;
    const float dn = denom[n * HEADS + (f >> 4)];
    const float y  = prev[n * FEAT + f] + agg[n * FEAT + f] / dn + bias[f];
    Y[n * FEAT + f] = y;
    s += y; sq += y * y;
  }
  atomicAdd(&fsum[f], s);
  atomicAdd(&fsq[f], sq);
}

// Kernel 5: fold mean/var/gamma/beta into per-feature scale/shift
__global__ void stats(const float* __restrict__ fsum, const float* __restrict__ fsq,
                      const float* __restrict__ gamma, const float* __restrict__ beta,
                      float* __restrict__ scale, float* __restrict__ shift)
{
  int f = threadIdx.x;
  float mu  = fsum[f] * (1.0f / (float)NODES);
  float var = fsq[f] * (1.0f / (float)NODES) - mu * mu;
  float inv = rsqrtf(var + LN_EPS);
  float sc  = gamma[f] * inv;
  scale[f] = sc;
  shift[f] = beta[f] - mu * sc;
}

// Kernel 6: in-place normalize + ELU
__global__ void norm_elu(float* Y, const float* __restrict__ scale,
                         const float* __restrict__ shift)
{
  int idx = blockIdx.x * blockDim.x + threadIdx.x;
  if (idx >= NODES * FEAT) return;
  int f = idx & (FEAT - 1);
  float v = Y[idx] * scale[f] + shift[f];
  Y[idx] = (v > 0.0f) ? v : (__expf(v) - 1.0f);
}

// Kernel 7: atomic mean-pool accumulation per graph
__global__ void pool_sum(const float* __restrict__ Hf, const int* __restrict__ batch,
                         float* __restrict__ pool, float* __restrict__ cnt)
{
  int idx = blockIdx.x * blockDim.x + threadIdx.x;
  if (idx >= NODES * FEAT) return;
  int n = idx >> 7, f = idx & (FEAT - 1);
  int b = batch[n];
  atomicAdd(&pool[b * FEAT + f], Hf[idx]);
  if (f == 0) atomicAdd(&cnt[b], 1.0f);
}

// Kernel 8: (pool/cnt) @ ro_w + ro_b -> 64x2 output
__global__ void readout(const float* __restrict__ pool, const float* __restrict__ cnt,
                        const float* __restrict__ ro_w, const float* __restrict__ ro_b,
                        float* __restrict__ out)
{
  int t = threadIdx.x;       // 0..127
  int g = t >> 1, j = t & 1;
  float c = fmaxf(cnt[g], 1.0f);
  float acc = 0.0f;
  #pragma unroll 8
  for (int f = 0; f < FEAT; ++f) acc += pool[g * FEAT + f] * ro_w[f * 2 + j];
  out[g * 2 + j] = acc / c + ro_b[j];
}

// ---------------------------------------------------------------------------
extern "C" void kernel_launch(void* const* d_in, const int* in_sizes, int n_in,
                              void* d_out, int out_size, void* d_ws, size_t ws_size,
                              hipStream_t stream)
{
  (void)in_sizes; (void)n_in; (void)out_size; (void)ws_size;
  const float* x     = (const float*)d_in[0];
  const int*   ei    = (const int*)d_in[1];
  const int*   batch = (const int*)d_in[2];
  const float* W     = (const float*)d_in[3];
  const float* asrc  = (const float*)d_in[4];
  const float* adst  = (const float*)d_in[5];
  const float* bias  = (const float*)d_in[6];
  const float* gamma = (const float*)d_in[7];
  const float* beta  = (const float*)d_in[8];
  const float* row   = (const float*)d_in[9];
  const float* rob   = (const float*)d_in[10];
  float* out = (float*)d_out;

  char* p = (char*)d_ws;
  auto carve = [&](size_t bytes) -> char* {
    char* q = p; p += (bytes + 255) & ~(size_t)255; return q;
  };
  float*    XP    = (float*)carve(sizeof(float) * NODES * FEAT);
  float*    AGG   = (float*)carve(sizeof(float) * NODES * FEAT);
  float*    Y0    = (float*)carve(sizeof(float) * NODES * FEAT);
  float*    Y1    = (float*)carve(sizeof(float) * NODES * FEAT);
  float*    AS    = (float*)carve(sizeof(float) * NODES * HEADS);
  float*    AD    = (float*)carve(sizeof(float) * NODES * HEADS);
  unsigned* MK    = (unsigned*)carve(sizeof(unsigned) * NODES * HEADS);
  float*    DEN   = (float*)carve(sizeof(float) * NODES * HEADS);
  float*    FSUM  = (float*)carve(sizeof(float) * FEAT);
  float*    FSQ   = (float*)carve(sizeof(float) * FEAT);
  float*    SCALE = (float*)carve(sizeof(float) * FEAT);
  float*    SHIFT = (float*)carve(sizeof(float) * FEAT);
  float*    POOL  = (float*)carve(sizeof(float) * NGRAPH * FEAT);
  float*    CNT   = (float*)carve(sizeof(float) * NGRAPH);

  hipMemsetAsync(Y1, 0, sizeof(float) * NODES * FEAT, stream);  // prev for layer 0

  // reference chain: prev=0,h=x -> out0 ; prev=x,h=out0 -> out1 ; prev=out0,h=out1 -> out2
  struct Plan { const float* h; const float* prev; float* out; };
  const Plan plan[3] = {
    { x,  Y1, Y0 },   // out0 -> Y0
    { Y0, x,  Y1 },   // out1 -> Y1
    { Y1, Y0, Y0 },   // out2 -> Y0 (prev read-before-write per element: safe)
  };

  for (int i = 0; i < 3; ++i) {
    const float* Wi = W + (size_t)i * FEAT * FEAT;
    hipMemsetAsync(DEN,  0, sizeof(float) * NODES * HEADS, stream);
    hipMemsetAsync(AGG,  0, sizeof(float) * NODES * FEAT,  stream);
    hipMemsetAsync(FSUM, 0, sizeof(float) * FEAT,          stream);
    hipMemsetAsync(FSQ,  0, sizeof(float) * FEAT,          stream);
    fill_u32<<<(NODES * HEADS + 255) / 256, 256, 0, stream>>>(MK, 0x007FFFFFu, NODES * HEADS);

    gemm_attn<<<(NODES + 15) / 16, 256, 0, stream>>>(
        plan[i].h, Wi, asrc + i * HEADS * CH, adst + i * HEADS * CH, XP, AS, AD);
    edge_max<<<(NEDGE + 255) / 256, 256, 0, stream>>>(ei, AS, AD, MK);
    edge_scatter<<<(NEDGE + 7) / 8, 256, 0, stream>>>(ei, AS, AD, MK, XP, DEN, AGG);
    node_update<<<(NODES + NPB - 1) / NPB, 128, 0, stream>>>(
        AGG, DEN, plan[i].prev, bias + i * FEAT, plan[i].out, FSUM, FSQ);
    stats<<<1, 128, 0, stream>>>(FSUM, FSQ, gamma + i * FEAT, beta + i * FEAT, SCALE, SHIFT);
    norm_elu<<<(NODES * FEAT + 255) / 256, 256, 0, stream>>>(plan[i].out, SCALE, SHIFT);
  }

  hipMemsetAsync(POOL, 0, sizeof(float) * NGRAPH * FEAT, stream);
  hipMemsetAsync(CNT,  0, sizeof(float) * NGRAPH,        stream);
  pool_sum<<<(NODES * FEAT + 255) / 256, 256, 0, stream>>>(Y0, batch, POOL, CNT);
  readout<<<1, 128, 0, stream>>>(POOL, CNT, row, rob, out);
}